// CARAFEplusplus_19000935317566
// MI455X (gfx1250) — compile-verified
//
#include <hip/hip_runtime.h>
#include <hip/hip_bf16.h>
#include <math.h>

// CARAFE++ 2x upsampling for MI455X (gfx1250, wave32).
// conv1x1/conv3x3 as fp32 WMMA GEMMs (V_WMMA_F32_16X16X4_F32), conv3x3 input
// staged to LDS via global_load_async_to_lds_b128 (ASYNCcnt), reassembly uses
// an LDS-broadcast kernel tile to kill redundant L2 traffic.

typedef __attribute__((ext_vector_type(2))) float v2f;
typedef __attribute__((ext_vector_type(8))) float v8f;

#define N_    4
#define CIN   256
#define CMID  64
#define HW    4096      // 64*64
#define KCH   100       // S*S*KK
#define KK_   25

// workspace layout in floats
#define WS_COMP 0
#define WS_RAW  (N_ * CMID * HW)                 // 1,048,576
#define WS_KUNF (WS_RAW + N_ * KCH * HW)         // 2,686,976 (16B-aligned)

// ---------------------------------------------------------------------------
// Kernel A: 1x1 conv (256 -> 64). One wave owns a 16(co) x 64(px) tile:
// one A fragment drives 4 WMMAs -> A traffic /4.
// ---------------------------------------------------------------------------
__global__ __launch_bounds__(256)
void carafe_conv1x1_wmma(const float* __restrict__ x,
                         const float* __restrict__ w_comp,
                         const float* __restrict__ b_comp,
                         float* __restrict__ comp) {
  const int wid   = blockIdx.x * 8 + (threadIdx.x >> 5);   // 1024 waves
  const int lane  = threadIdx.x & 31;
  const int n     = wid >> 8;                  // 4*64 wave-tiles per image
  const int r     = wid & 255;
  const int co0   = (r >> 6) << 4;             // 0,16,32,48
  const int p0    = (r & 63) << 6;             // 64-px strip
  const int row   = lane & 15;                 // A: M row / B,D: N col
  const int khalf = lane >> 4;

  const float* Wp = w_comp + (size_t)(co0 + row) * CIN;
  const float* Xp = x + (size_t)n * CIN * HW + p0 + row;

  v8f acc0 = {}, acc1 = {}, acc2 = {}, acc3 = {};
#pragma unroll 2
  for (int k = 0; k < CIN; k += 4) {
    const int kb = k + khalf * 2;
    v2f a;
    a.x = Wp[kb];
    a.y = Wp[kb + 1];
    const float* xb0 = Xp + (size_t)kb * HW;
    const float* xb1 = Xp + (size_t)(kb + 1) * HW;
    v2f b0, b1, b2, b3;
    b0.x = xb0[0];  b0.y = xb1[0];
    b1.x = xb0[16]; b1.y = xb1[16];
    b2.x = xb0[32]; b2.y = xb1[32];
    b3.x = xb0[48]; b3.y = xb1[48];
    if (k + 8 < CIN)
      __builtin_prefetch(xb0 + (size_t)8 * HW, 0, 1);
    acc0 = __builtin_amdgcn_wmma_f32_16x16x4_f32(false, a, false, b0, (short)0, acc0, false, false);
    acc1 = __builtin_amdgcn_wmma_f32_16x16x4_f32(false, a, false, b1, (short)0, acc1, false, false);
    acc2 = __builtin_amdgcn_wmma_f32_16x16x4_f32(false, a, false, b2, (short)0, acc2, false, false);
    acc3 = __builtin_amdgcn_wmma_f32_16x16x4_f32(false, a, false, b3, (short)0, acc3, false, false);
  }

  const int ncol  = lane & 15;
  const int mbase = khalf * 8;
#pragma unroll
  for (int v = 0; v < 8; ++v) {
    const int m   = mbase + v;
    const float bia = b_comp[co0 + m];
    float* o = comp + ((size_t)n * CMID + co0 + m) * HW + p0 + ncol;
    o[0]  = acc0[v] + bia;
    o[16] = acc1[v] + bia;
    o[32] = acc2[v] + bia;
    o[48] = acc3[v] + bia;
  }
}

// ---------------------------------------------------------------------------
// Kernel B: 3x3 conv (64 -> 100). Block = one (n, h) row of 64 px.
// Input slab (64 ci x 3 rows x 64 w, halo cols zeroed, 72-float padded rows,
// 54 KB LDS) staged via async copy engine; waves read B from LDS, A from L2.
// Each wave holds 3-4 M-tile accumulators so one B fragment feeds 3-4 WMMAs.
// ---------------------------------------------------------------------------
__global__ __launch_bounds__(256)
void carafe_conv3x3_wmma(const float* __restrict__ comp,
                         const float* __restrict__ w_enc,
                         float* __restrict__ raw) {
  __shared__ float lds[64 * 3 * 72];           // 55,296 B
  const int n   = blockIdx.x >> 6;
  const int h0  = blockIdx.x & 63;
  const int tid = threadIdx.x;

  // zero halo columns: 2 float4 pads per (ci,row) slab row (384 total)
#pragma unroll
  for (int j = 0; j < 2; ++j) {
    const int pp = tid + j * 256;
    if (pp < 384) {
      const int rowp = pp >> 1, side = pp & 1;
      float4 z = {0.f, 0.f, 0.f, 0.f};
      *(float4*)&lds[rowp * 72 + side * 68] = z;
    }
  }

  // interior: 192 slab rows x 16 float4, async global->LDS (no VGPR bounce)
  const float* cb = comp + (size_t)n * CMID * HW;
#pragma unroll
  for (int i = 0; i < 12; ++i) {
    const int e    = i * 256 + tid;
    const int rowi = e >> 4;                   // ci*3 + r
    const int w4   = e & 15;
    const int ci   = rowi / 3;
    const int rr   = rowi - ci * 3;
    const int hh   = h0 - 1 + rr;
    const unsigned ldsoff = (unsigned)(rowi * 72 + 4 + w4 * 4) * 4u;
    if ((unsigned)hh < 64u) {
      const float* src = cb + (size_t)ci * HW + hh * 64 + w4 * 4;
      const unsigned dst = (unsigned)(uintptr_t)&lds[0] + ldsoff;
      asm volatile("global_load_async_to_lds_b128 %0, %1, off"
                   :: "v"(dst), "v"(src) : "memory");
    } else {
      float4 z = {0.f, 0.f, 0.f, 0.f};
      *(float4*)&lds[rowi * 72 + 4 + w4 * 4] = z;
    }
  }
  asm volatile("s_wait_asynccnt 0x0" ::: "memory");
  __syncthreads();

  const int wave  = tid >> 5;
  const int lane  = tid & 31;
  const int nt    = wave & 3;                  // 16-px column tile
  const int mbase  = __builtin_amdgcn_readfirstlane((wave >> 2) ? 4 : 0);
  const int mcount = __builtin_amdgcn_readfirstlane((wave >> 2) ? 3 : 4);
  const int row   = lane & 15;
  const int khalf = lane >> 4;
  const int wcol  = nt * 16 + row;             // w coordinate of this lane's N col

  v8f acc[4] = {v8f{}, v8f{}, v8f{}, v8f{}};
  for (int k = 0; k < 576; k += 4) {
    v2f bfrag;
#pragma unroll
    for (int v = 0; v < 2; ++v) {
      const int kidx = k + khalf * 2 + v;
      const int ci   = kidx / 9;
      const int rr9  = kidx - ci * 9;
      const int kh   = rr9 / 3;
      const int kw   = rr9 - kh * 3;
      const float bv = lds[(ci * 3 + kh) * 72 + wcol + kw + 3];
      if (v == 0) bfrag.x = bv; else bfrag.y = bv;
    }
#pragma unroll
    for (int mi = 0; mi < 4; ++mi) {
      if (mi < mcount) {                       // uniform (readfirstlane) branch
        const int co     = (mbase + mi) * 16 + row;
        const int cosafe = co < KCH ? co : (KCH - 1);
        const float* Ap  = w_enc + (size_t)cosafe * 576 + k + khalf * 2;
        v2f a;
        a.x = Ap[0];
        a.y = Ap[1];
        if (co >= KCH) { a.x = 0.f; a.y = 0.f; }
        acc[mi] = __builtin_amdgcn_wmma_f32_16x16x4_f32(false, a, false, bfrag,
                                                        (short)0, acc[mi], false, false);
      }
    }
  }

  const int ncol = lane & 15;
  const int p    = h0 * 64 + nt * 16 + ncol;
#pragma unroll
  for (int mi = 0; mi < 4; ++mi) {
    if (mi < mcount) {
#pragma unroll
      for (int v = 0; v < 8; ++v) {
        const int m = (mbase + mi) * 16 + khalf * 8 + v;
        if (m < KCH)
          raw[((size_t)n * KCH + m) * HW + p] = acc[mi][v];
      }
    }
  }
}

// ---------------------------------------------------------------------------
// Kernel C: pixel-shuffle view + softmax over 25 weights, regrouped to
// [n][pixel][kk][q] so kernel D can load float4 per kk.
// ---------------------------------------------------------------------------
__global__ __launch_bounds__(256)
void carafe_softmax_regroup(const float* __restrict__ raw,
                            const float* __restrict__ b_enc,
                            float* __restrict__ kunf) {
  const int t = blockIdx.x * 256 + threadIdx.x;   // 65536 threads
  const int q = t & 3;
  const int p = (t >> 2) & 4095;
  const int n = t >> 14;

  float v[KK_];
  float mx = -3.0e38f;
#pragma unroll
  for (int kk = 0; kk < KK_; ++kk) {
    v[kk] = raw[((size_t)n * KCH + kk * 4 + q) * HW + p] + b_enc[kk * 4 + q];
    mx = fmaxf(mx, v[kk]);
  }
  float s = 0.0f;
#pragma unroll
  for (int kk = 0; kk < KK_; ++kk) {
    v[kk] = __expf(v[kk] - mx);
    s += v[kk];
  }
  const float inv = 1.0f / s;
  float* o = kunf + (((size_t)n * HW + p) * KK_) * 4 + q;
#pragma unroll
  for (int kk = 0; kk < KK_; ++kk)
    o[kk * 4] = v[kk] * inv;
}

// ---------------------------------------------------------------------------
// Kernel D: content-aware reassembly + pixel shuffle.
// Block = 16 channels x 16 pixels; the 16px x 25 float4 kernel tile lives in
// LDS and is broadcast across channel lanes (kunf L2 traffic / 16).
// ---------------------------------------------------------------------------
__global__ __launch_bounds__(256)
void carafe_reassemble(const float* __restrict__ x,
                       const float* __restrict__ kunf,
                       float* __restrict__ out) {
  __shared__ float4 kt[16 * KK_];              // 6.4 KB
  const int b  = blockIdx.x;                   // 16384 blocks
  const int ct = b & 15;
  const int wt = (b >> 4) & 3;
  const int h  = (b >> 6) & 63;
  const int n  = b >> 12;
  const int tid = threadIdx.x;

  const float4* ks = (const float4*)kunf + ((size_t)n * HW + h * 64 + wt * 16) * KK_;
  for (int e = tid; e < 16 * KK_; e += 256)
    kt[e] = ks[e];
  __syncthreads();

  const int wl = tid & 15, cl = tid >> 4;
  const int c = ct * 16 + cl;
  const int w = wt * 16 + wl;
  const float* xc = x + ((size_t)n * CIN + c) * HW;

  float4 acc = {0.f, 0.f, 0.f, 0.f};
#pragma unroll
  for (int i = 0; i < 5; ++i) {
    const int ih = h + i - 2;
    const bool okh = (unsigned)ih < 64u;
#pragma unroll
    for (int j = 0; j < 5; ++j) {
      const int iw = w + j - 2;
      float xv = 0.0f;
      if (okh && (unsigned)iw < 64u)
        xv = xc[ih * 64 + iw];
      const float4 kv = kt[wl * KK_ + (i * 5 + j)];
      acc.x = fmaf(xv, kv.x, acc.x);
      acc.y = fmaf(xv, kv.y, acc.y);
      acc.z = fmaf(xv, kv.z, acc.z);
      acc.w = fmaf(xv, kv.w, acc.w);
    }
  }

  float* ob = out + ((size_t)n * CIN + c) * (128 * 128) + (size_t)(2 * h) * 128 + 2 * w;
  float2 r0 = {acc.x, acc.y};
  float2 r1 = {acc.z, acc.w};
  *(float2*)(ob)       = r0;
  *(float2*)(ob + 128) = r1;
}

// ---------------------------------------------------------------------------
extern "C" void kernel_launch(void* const* d_in, const int* in_sizes, int n_in,
                              void* d_out, int out_size, void* d_ws, size_t ws_size,
                              hipStream_t stream) {
  (void)in_sizes; (void)n_in; (void)out_size; (void)ws_size;
  const float* x      = (const float*)d_in[0];
  const float* w_comp = (const float*)d_in[1];
  const float* b_comp = (const float*)d_in[2];
  const float* w_enc  = (const float*)d_in[3];
  const float* b_enc  = (const float*)d_in[4];
  float* out = (float*)d_out;
  float* ws  = (float*)d_ws;

  float* comp = ws + WS_COMP;   // (4, 64, 64, 64)
  float* raw  = ws + WS_RAW;    // (4, 100, 64, 64)
  float* kunf = ws + WS_KUNF;   // (4, 4096, 25, 4)

  carafe_conv1x1_wmma<<<128, 256, 0, stream>>>(x, w_comp, b_comp, comp);
  carafe_conv3x3_wmma<<<256, 256, 0, stream>>>(comp, w_enc, raw);
  carafe_softmax_regroup<<<256, 256, 0, stream>>>(raw, b_enc, kunf);
  carafe_reassemble<<<16384, 256, 0, stream>>>(x, kunf, out);
}